// DualAttentionCriticNetwork_35562329211478
// MI455X (gfx1250) — compile-verified
//
#include <hip/hip_runtime.h>
#include <hip/hip_bf16.h>
#include <math.h>

// ---------------------------------------------------------------------------
// DualAttentionCriticNetwork for MI455X (gfx1250), fp32 end-to-end.
// Memory-bound on the 256MB noise tensor (~11.5us @ 23.3 TB/s); attention
// math done with V_WMMA_F32_16X16X4_F32 (full-precision CDNA5 WMMA).
// Noise is streamed with non-temporal b128 loads so L2 (192MB) keeps the hot
// working set (W1, workspace, w_oa) instead of the 256MB one-shot stream.
// ---------------------------------------------------------------------------

typedef __attribute__((ext_vector_type(2))) float v2f;
typedef __attribute__((ext_vector_type(4))) float v4f;
typedef __attribute__((ext_vector_type(8))) float v8f;

#define B_    16
#define N_    32
#define DOBS  128
#define NACT  16
#define DOA   144
#define DOUT  128
#define HID   64

// workspace layout (float offsets)
// proj p in {0:K_oa, 1:Q_oa, 2:AVa, 3:AVp, 4:Ko, 5:Qo, 6:AVO}, each [B][32][128]
#define WS_PROJ  0
#define WS_S     (7 * B_ * N_ * DOUT)            // S = w_oa @ AVa, [B][32][128]
#define WS_WFO   (WS_S  + B_ * N_ * DOUT)        // wfo (already /32), [B][32][128]
#define WS_DAV   (WS_WFO + B_ * N_ * DOUT)       // AVp - AVa, [B][32][128]

// d_out layout: value[16384] | w_oa[16384] | w_o[16384]
#define OUT_WOA  (B_ * N_ * N_)
#define OUT_WO   (2 * B_ * N_ * N_)

// ---------------------------------------------------------------------------
// Kernel 1: seven [32,Kin] x [Kin,128] projections per batch via f32 WMMA.
// Fragment layout (ISA 7.12.2): l=lane&15, half=lane>>4.
//   A(16x4): a.x=A[l][2h], a.y=A[l][2h+1]
//   B(4x16): b.x=B[2h][l],  b.y=B[2h+1][l]   (= W row-major consecutive pair)
//   C(16x16): c[r] = C[half*8+r][l]
// ---------------------------------------------------------------------------
__global__ void __launch_bounds__(128)
proj_kernel(const float* __restrict__ states,
            const float* __restrict__ policies,
            const float* __restrict__ actions,
            const float* __restrict__ Wk_oa, const float* __restrict__ Wq_oa,
            const float* __restrict__ Wv_oa, const float* __restrict__ Wk_o,
            const float* __restrict__ Wq_o,  const float* __restrict__ Wv_o,
            float* __restrict__ ws)
{
    __shared__ float X[N_][DOA];
    const int p   = blockIdx.x % 7;
    const int b   = blockIdx.x / 7;
    const int tid = threadIdx.x;
    const int kin = (p < 4) ? DOA : DOBS;

    // stage input rows into LDS (states + actions/policies tail)
    for (int idx = tid; idx < N_ * kin; idx += blockDim.x) {
        const int m = idx / kin, c = idx % kin;
        float v;
        if (c < DOBS) {
            v = states[(size_t)(b * N_ + m) * DOBS + c];
        } else {
            const float* tail = (p == 3) ? policies : actions;
            v = tail[(size_t)(b * N_ + m) * NACT + (c - DOBS)];
        }
        X[m][c] = v;
    }
    __syncthreads();

    const float* W;
    switch (p) {
        case 0: W = Wk_oa; break;
        case 1: W = Wq_oa; break;
        case 2: case 3: W = Wv_oa; break;
        case 4: W = Wk_o; break;
        case 5: W = Wq_o; break;
        default: W = Wv_o; break;
    }

    const int wave = tid >> 5;
    const int lane = tid & 31;
    const int l = lane & 15, half = lane >> 4;
    float* outp = ws + WS_PROJ + (size_t)(p * B_ + b) * (N_ * DOUT);

    // 16 output tiles (2 x 8), 4 waves -> 4 tiles each
    for (int t = wave; t < 16; t += 4) {
        const int mt = t >> 3, nt = t & 7;
        v8f c = {};
        const float* wrow = W + (size_t)(nt * 16 + l) * kin + 2 * half;
        const float* xrow = &X[mt * 16 + l][2 * half];
        for (int k0 = 0; k0 < kin; k0 += 4) {
            v2f a;  a.x  = xrow[k0];  a.y  = xrow[k0 + 1];
            v2f bb; bb.x = wrow[k0];  bb.y = wrow[k0 + 1];
            c = __builtin_amdgcn_wmma_f32_16x16x4_f32(false, a, false, bb,
                                                      (short)0, c, false, false);
        }
        const int row = mt * 16 + half * 8;
        #pragma unroll
        for (int r = 0; r < 8; ++r)
            outp[(size_t)(row + r) * DOUT + nt * 16 + l] = c[r];
    }
}

// ---------------------------------------------------------------------------
// Kernel 2 (one block per batch, 8 waves):
//   scores = K @ Q^T (WMMA), softmax rows -> w_oa / w_o (written to d_out),
//   S = w_oa @ AVa, wfo = (w_o @ AVO)/32 (WMMA), dAV = AVp - AVa.
// ---------------------------------------------------------------------------
__global__ void __launch_bounds__(256)
attn_kernel(float* __restrict__ ws, float* __restrict__ out)
{
    __shared__ float sc[2][N_][N_];
    const int b    = blockIdx.x;
    const int tid  = threadIdx.x;
    const int wave = tid >> 5;
    const int lane = tid & 31;
    const int l = lane & 15, half = lane >> 4;

    const float* K_oa = ws + WS_PROJ + (size_t)(0 * B_ + b) * (N_ * DOUT);
    const float* Q_oa = ws + WS_PROJ + (size_t)(1 * B_ + b) * (N_ * DOUT);
    const float* AVa  = ws + WS_PROJ + (size_t)(2 * B_ + b) * (N_ * DOUT);
    const float* AVp  = ws + WS_PROJ + (size_t)(3 * B_ + b) * (N_ * DOUT);
    const float* Ko   = ws + WS_PROJ + (size_t)(4 * B_ + b) * (N_ * DOUT);
    const float* Qo   = ws + WS_PROJ + (size_t)(5 * B_ + b) * (N_ * DOUT);
    const float* AVO  = ws + WS_PROJ + (size_t)(6 * B_ + b) * (N_ * DOUT);

    // dAV = AVp - AVa
    {
        float* dav = ws + WS_DAV + (size_t)b * (N_ * DOUT);
        for (int i = tid; i < N_ * DOUT; i += 256) dav[i] = AVp[i] - AVa[i];
    }

    // scores: 2 branches x (2x2 tiles) = 8 tiles, one per wave.
    // T[j][i] = K[j] . Q[i]  (row-major x row-major^T)
    {
        const int br = wave >> 2;          // 0: obs-action, 1: obs-only
        const int t  = wave & 3;
        const int mt = t >> 1, nt = t & 1;
        const float* Am = br ? Ko : K_oa;
        const float* Bm = br ? Qo : Q_oa;
        v8f c = {};
        const float* arow = Am + (size_t)(mt * 16 + l) * DOUT + 2 * half;
        const float* brow = Bm + (size_t)(nt * 16 + l) * DOUT + 2 * half;
        for (int k0 = 0; k0 < DOUT; k0 += 4) {
            v2f a;  a.x  = arow[k0];  a.y  = arow[k0 + 1];
            v2f bb; bb.x = brow[k0];  bb.y = brow[k0 + 1];
            c = __builtin_amdgcn_wmma_f32_16x16x4_f32(false, a, false, bb,
                                                      (short)0, c, false, false);
        }
        const int row = mt * 16 + half * 8;
        #pragma unroll
        for (int r = 0; r < 8; ++r) sc[br][row + r][nt * 16 + l] = c[r];
    }
    __syncthreads();

    // row softmax (scale 1/sqrt(128)); one thread per row; write outputs
    if (tid < 64) {
        const int br = tid >> 5, row = tid & 31;
        float* r = sc[br][row];
        const float scale = 0.08838834764831844f;  // 1/sqrt(128)
        float mx = -3.0e38f;
        for (int i = 0; i < N_; ++i) mx = fmaxf(mx, r[i] * scale);
        float s = 0.f;
        for (int i = 0; i < N_; ++i) { float e = __expf(r[i] * scale - mx); r[i] = e; s += e; }
        const float inv = 1.0f / s;
        float* od = out + (br ? OUT_WO : OUT_WOA) + (size_t)b * (N_ * N_) + row * N_;
        for (int i = 0; i < N_; ++i) { r[i] *= inv; od[i] = r[i]; }
    }
    __syncthreads();

    // S = w_oa @ AVa (raw), wfo = (w_o @ AVO)/32.  32 tiles over 8 waves.
    for (int t = wave; t < 32; t += 8) {
        const int br = t >> 4;
        const int tt = t & 15;
        const int mt = tt >> 3, nt = tt & 7;
        const float* Bg = br ? AVO : AVa;
        v8f c = {};
        for (int k0 = 0; k0 < N_; k0 += 4) {
            v2f a;  a.x  = sc[br][mt * 16 + l][k0 + 2 * half];
                    a.y  = sc[br][mt * 16 + l][k0 + 2 * half + 1];
            v2f bb; bb.x = Bg[(size_t)(k0 + 2 * half)     * DOUT + nt * 16 + l];
                    bb.y = Bg[(size_t)(k0 + 2 * half + 1) * DOUT + nt * 16 + l];
            c = __builtin_amdgcn_wmma_f32_16x16x4_f32(false, a, false, bb,
                                                      (short)0, c, false, false);
        }
        float* dst = ws + (br ? WS_WFO : WS_S) + (size_t)b * (N_ * DOUT);
        const float mul = br ? (1.0f / 32.0f) : 1.0f;
        const int row = mt * 16 + half * 8;
        #pragma unroll
        for (int r = 0; r < 8; ++r)
            dst[(size_t)(row + r) * DOUT + nt * 16 + l] = c[r] * mul;
    }
}

// ---------------------------------------------------------------------------
// Kernel 3 (bandwidth kernel, 16384 blocks x 128 threads): one (b,k,j) each.
// Streams the 16KB noise tile with non-temporal global_load_b128 (8 per
// thread), LDS-reduces the i-partitions, forms wav, runs the 256->64->1 MLP.
// ---------------------------------------------------------------------------
__global__ void __launch_bounds__(128)
final_kernel(const float* __restrict__ noise,
             const float* __restrict__ W1, const float* __restrict__ W2,
             const float* __restrict__ ws, float* __restrict__ out)
{
    __shared__ v4f  red[4][N_];        // per-ipartition column sums (16B aligned)
    __shared__ v4f  node4[2 * DOUT / 4]; // [wfo | wav] as float4 groups
    __shared__ float hred[HID];
    const int bkj = blockIdx.x;
    const int j = bkj & 31, k = (bkj >> 5) & 31, b = bkj >> 10;
    const int t  = threadIdx.x;
    const int dg = t & 31;             // float4 group along d (d = 4*dg..4*dg+3)
    const int ip = t >> 5;             // i partition (0..3)

    // --- noise column sums: thread sums i = ip, ip+4, ..., ip+28 ------------
    const v4f* np = (const v4f*)(noise + (size_t)bkj * (N_ * DOUT)) + dg;
    v4f acc = (v4f)0.0f;
    #pragma unroll
    for (int it = 0; it < 8; ++it)
        acc += __builtin_nontemporal_load(np + (size_t)(ip + 4 * it) * (DOUT / 4));
    red[ip][dg] = acc;
    __syncthreads();

    // --- wave 0: finish mean, assemble node[256] ---------------------------
    if (t < N_) {
        const v4f sum = (red[0][t] + red[1][t]) + (red[2][t] + red[3][t]);
        // mean((noise-0.5)*0.1, axis=i) = (mean(noise)-0.5)*0.1
        const float inv32 = 1.0f / 32.0f;
        const v4f nm = (sum * inv32 - 0.5f) * 0.1f;

        const float woa = out[OUT_WOA + (size_t)b * (N_ * N_) + k * N_ + j];
        const v4f sp = ((const v4f*)(ws + WS_S   + (size_t)b * (N_ * DOUT) + k * DOUT))[t];
        const v4f dv = ((const v4f*)(ws + WS_DAV + (size_t)b * (N_ * DOUT) + j * DOUT))[t];
        const v4f wf = ((const v4f*)(ws + WS_WFO + (size_t)b * (N_ * DOUT) + k * DOUT))[t];

        node4[t]            = wf;                            // wfo (already /32)
        node4[DOUT / 4 + t] = (sp + woa * dv) * inv32 + nm;  // wav
    }
    __syncthreads();

    // --- MLP: h = leaky_relu(node @ W1^T), value = h . W2 ------------------
    if (t < HID) {
        const v4f* w1 = (const v4f*)(W1 + (size_t)t * (2 * DOUT));
        float h = 0.f;
        #pragma unroll 8
        for (int c = 0; c < 2 * DOUT / 4; ++c) {
            const v4f pr = node4[c] * w1[c];
            h += (pr.x + pr.y) + (pr.z + pr.w);
        }
        h = (h > 0.f) ? h : 0.01f * h;           // leaky_relu(0.01)
        hred[t] = h * W2[t];
    }
    __syncthreads();
    if (t == 0) {
        float v = 0.f;
        #pragma unroll
        for (int c = 0; c < HID; ++c) v += hred[c];
        out[bkj] = v;
    }
}

// ---------------------------------------------------------------------------
extern "C" void kernel_launch(void* const* d_in, const int* in_sizes, int n_in,
                              void* d_out, int out_size, void* d_ws, size_t ws_size,
                              hipStream_t stream) {
    const float* states   = (const float*)d_in[0];
    const float* policies = (const float*)d_in[1];
    const float* actions  = (const float*)d_in[2];
    const float* noise    = (const float*)d_in[3];
    const float* Wk_oa    = (const float*)d_in[4];
    const float* Wq_oa    = (const float*)d_in[5];
    const float* Wv_oa    = (const float*)d_in[6];
    const float* Wk_o     = (const float*)d_in[7];
    const float* Wq_o     = (const float*)d_in[8];
    const float* Wv_o     = (const float*)d_in[9];
    const float* W1       = (const float*)d_in[10];
    const float* W2       = (const float*)d_in[11];
    float* ws  = (float*)d_ws;
    float* out = (float*)d_out;

    proj_kernel<<<dim3(B_ * 7), dim3(128), 0, stream>>>(
        states, policies, actions, Wk_oa, Wq_oa, Wv_oa, Wk_o, Wq_o, Wv_o, ws);
    attn_kernel<<<dim3(B_), dim3(256), 0, stream>>>(ws, out);
    final_kernel<<<dim3(B_ * N_ * N_), dim3(128), 0, stream>>>(
        noise, W1, W2, ws, out);
}